// GCNLayer_71296457114204
// MI455X (gfx1250) — compile-verified
//
#include <hip/hip_runtime.h>
#include <cstdint>

// ---------------------------------------------------------------------------
// GCN layer: out = A @ embeds, COO with edge_row sorted ascending.
// Row-tiled WMMA formulation: each wave owns a 16-row output tile held in
// four v_wmma_f32_16x16x4_f32 accumulators (16x64 f32). Edges for a tile are
// a contiguous slice of the sorted edge list (binary search). Edge stream is
// double-buffered into LDS with global_load_async_to_lds_b128 (ASYNCcnt).
// No atomics, no zero-init pass: every output element stored exactly once.
// ---------------------------------------------------------------------------

typedef __attribute__((ext_vector_type(2))) float v2f;
typedef __attribute__((ext_vector_type(8))) float v8f;

#define WPB    8      // waves per block (256 threads)
#define CHUNK  128    // edges staged per async batch (per wave)
#define DIM    64     // embedding dim (fixed by reference)
#define TILE   16     // output rows per wave (WMMA M)

struct __align__(16) EdgeBuf {
    int   rows[CHUNK];
    int   cols[CHUNK];
    float vals[CHUNK];
};

__device__ __forceinline__ int imin(int a, int b) { return a < b ? a : b; }

// Low 32 bits of a flat LDS pointer are the LDS byte address (ISA 10.2).
__device__ __forceinline__ uint32_t lds_addr32(const void* p) {
    return (uint32_t)(uintptr_t)p;
}

// 16B async copy global -> LDS, tracked by ASYNCcnt.
__device__ __forceinline__ void async_cp16(uint32_t lds, const void* gaddr) {
    asm volatile("global_load_async_to_lds_b128 %0, %1, off"
                 :: "v"(lds), "v"(gaddr) : "memory");
}

__device__ __forceinline__ void stage_async(EdgeBuf* b,
                                            const int* er, const int* ec,
                                            const float* ev, int e0, int lane) {
    const int o = lane * 4;                       // 32 lanes x 4 elems = 128
    async_cp16(lds_addr32(&b->rows[o]), er + e0 + o);
    async_cp16(lds_addr32(&b->cols[o]), ec + e0 + o);
    async_cp16(lds_addr32(&b->vals[o]), ev + e0 + o);
}

// Tail path (partial chunk): plain loads + LDS stores, padded to a multiple
// of 4 with inert (row=-1, col=0, val=0) entries (val=0 => A entry = 0).
__device__ __forceinline__ void stage_sync(EdgeBuf* b,
                                           const int* er, const int* ec,
                                           const float* ev,
                                           int e0, int cnt, int lane) {
    const int cntp = (cnt + 3) & ~3;
    for (int j = lane; j < cntp; j += 32) {
        const bool in = (j < cnt);
        b->rows[j] = in ? er[e0 + j] : -1;
        b->cols[j] = in ? ec[e0 + j] : 0;
        b->vals[j] = in ? ev[e0 + j] : 0.0f;
    }
    asm volatile("s_wait_dscnt 0" ::: "memory");  // cross-lane LDS visibility
}

// First edge index whose row >= key (edge_row sorted ascending).
__device__ __forceinline__ int lower_bound_row(const int* __restrict__ er,
                                               int E, int key) {
    int lo = 0, hi = E;
    while (lo < hi) {
        const int mid = (lo + hi) >> 1;
        if (er[mid] < key) lo = mid + 1; else hi = mid;
    }
    return lo;
}

__global__ void __launch_bounds__(WPB * 32)
gcn_spmm_wmma_kernel(const int* __restrict__ er, const int* __restrict__ ec,
                     const float* __restrict__ ev, const float* __restrict__ emb,
                     float* __restrict__ out, int E, int N, int ntiles) {
    __shared__ EdgeBuf buf[WPB][2];

    const int lane = threadIdx.x & 31;
    // Force wave-uniformity so binary search / staging scalarize (s_load).
    const int wave = __builtin_amdgcn_readfirstlane(threadIdx.x >> 5);
    const int tile = blockIdx.x * WPB + wave;
    if (tile >= ntiles) return;                   // wave-uniform exit
    const int base = tile * TILE;

    // Contiguous edge slice for rows [base, base+16).
    const int s   = lower_bound_row(er, E, base);
    const int e   = lower_bound_row(er, E, base + TILE);
    const int cnt = e - s;
    const int nc  = (cnt + CHUNK - 1) / CHUNK;

    // Four 16x16 accumulators = full 16x64 output tile.
    v8f acc0 = {}, acc1 = {}, acc2 = {}, acc3 = {};

    const int n  = lane & 15;                     // WMMA N / M-within-half
    const int hi = lane >> 4;                     // half-wave select
    const int m  = base + n;                      // output row this lane tests

    if (nc > 0) {
        if (cnt >= CHUNK) stage_async(&buf[wave][0], er, ec, ev, s, lane);
        else              stage_sync (&buf[wave][0], er, ec, ev, s, cnt, lane);

        for (int k = 0; k < nc; ++k) {
            bool pre = false;
            if (k + 1 < nc) {
                const int c1n = imin(CHUNK, cnt - (k + 1) * CHUNK);
                const int e1  = s + (k + 1) * CHUNK;
                if (c1n == CHUNK) {
                    stage_async(&buf[wave][(k + 1) & 1], er, ec, ev, e1, lane);
                    pre = true;
                } else {
                    stage_sync(&buf[wave][(k + 1) & 1], er, ec, ev, e1, c1n, lane);
                }
            }
            if (pre) asm volatile("s_wait_asynccnt 3" ::: "memory");
            else     asm volatile("s_wait_asynccnt 0" ::: "memory");

            EdgeBuf* b = &buf[wave][k & 1];
            const int ck  = imin(CHUNK, cnt - k * CHUNK);
            const int ck4 = (ck + 3) & ~3;

            for (int i = 0; i < ck4; i += 4) {
                // Uniform-address wide LDS broadcasts of 4 edges.
                const int4   r = *(const int4*)  (&b->rows[i]);
                const int4   c = *(const int4*)  (&b->cols[i]);
                const float4 v = *(const float4*)(&b->vals[i]);

                // Lane's two K-slots: lanes 0-15 -> edges {i,i+1} (K=0,1),
                // lanes 16-31 -> edges {i+2,i+3} (K=2,3). (ISA 7.12.2)
                const int   colA = hi ? c.z : c.x;
                const int   colB = hi ? c.w : c.y;
                const int   rA   = hi ? r.z : r.x;
                const int   rB   = hi ? r.w : r.y;
                const float vA   = hi ? v.z : v.x;
                const float vB   = hi ? v.w : v.y;

                // B fragments: one dword gather per VGPR per column-chunk.
                const float* pA = emb + (size_t)colA * DIM + n;
                const float* pB = emb + (size_t)colB * DIM + n;
                const v2f b0 = { pA[ 0], pB[ 0] };
                const v2f b1 = { pA[16], pB[16] };
                const v2f b2 = { pA[32], pB[32] };
                const v2f b3 = { pA[48], pB[48] };

                // A fragment: select val where the edge's row hits lane's M.
                const v2f a = { (rA == m) ? vA : 0.0f,
                                (rB == m) ? vB : 0.0f };

                acc0 = __builtin_amdgcn_wmma_f32_16x16x4_f32(
                           false, a, false, b0, (short)0, acc0, false, false);
                acc1 = __builtin_amdgcn_wmma_f32_16x16x4_f32(
                           false, a, false, b1, (short)0, acc1, false, false);
                acc2 = __builtin_amdgcn_wmma_f32_16x16x4_f32(
                           false, a, false, b2, (short)0, acc2, false, false);
                acc3 = __builtin_amdgcn_wmma_f32_16x16x4_f32(
                           false, a, false, b3, (short)0, acc3, false, false);
            }
        }
    }

    // Store the 16x64 tile: C/D VGPR j holds rows (j, j+8) across half-waves.
    #pragma unroll
    for (int j = 0; j < 8; ++j) {
        const int row = base + j + 8 * hi;
        if (row < N) {
            const size_t o = (size_t)row * DIM + n;
            out[o +  0] = acc0[j];
            out[o + 16] = acc1[j];
            out[o + 32] = acc2[j];
            out[o + 48] = acc3[j];
        }
    }
}

extern "C" void kernel_launch(void* const* d_in, const int* in_sizes, int n_in,
                              void* d_out, int out_size, void* d_ws, size_t ws_size,
                              hipStream_t stream) {
    const int*   edge_row = (const int*)  d_in[0];
    const int*   edge_col = (const int*)  d_in[1];
    const float* edge_val = (const float*)d_in[2];
    const float* embeds   = (const float*)d_in[3];
    // d_in[4] = num_nodes (device scalar); N recovered from out_size = N*DIM.
    const int E = in_sizes[0];
    const int N = out_size / DIM;
    const int ntiles = (N + TILE - 1) / TILE;
    float* out = (float*)d_out;

    const int blocks = (ntiles + WPB - 1) / WPB;
    gcn_spmm_wmma_kernel<<<blocks, WPB * 32, 0, stream>>>(
        edge_row, edge_col, edge_val, embeds, out, E, N, ntiles);
}